// BayesianPDA_86397562127150
// MI455X (gfx1250) — compile-verified
//
#include <hip/hip_runtime.h>
#include <stdint.h>

// ---------------------------------------------------------------------------
// BayesianPDA: mu-DP (logaddexp recurrence over rows) + two-way softmax.
// W: [B, Na, Nb] f32, mask: [B, Na, Nb] u8 -> pi: [B, Na, Nb, 2] f32.
// One workgroup per batch; 1024 threads = one column each; DP row kept in LDS
// (double-buffered, one barrier per row). W rows staged ahead of the chain
// with gfx1250 async global->LDS copies tracked by ASYNCcnt.
// ---------------------------------------------------------------------------

typedef __attribute__((ext_vector_type(2))) float v2f;
typedef int v4i __attribute__((vector_size(16)));
// Builtin expects pointers to 16-byte int vectors: global (AS1) src, LDS (AS3) dst.
typedef __attribute__((address_space(1))) v4i* gv4i_p;
typedef __attribute__((address_space(3))) v4i* lv4i_p;

#ifndef __has_builtin
#define __has_builtin(x) 0
#endif

#if __has_builtin(__builtin_amdgcn_global_load_async_to_lds_b128)
#define HAVE_ASYNC_LDS 1
#else
#define HAVE_ASYNC_LDS 0
#endif

#if __has_builtin(__builtin_amdgcn_s_wait_asynccnt)
#define WAIT_ASYNC0() __builtin_amdgcn_s_wait_asynccnt(0)
#else
#define WAIT_ASYNC0() asm volatile("s_wait_asynccnt 0" ::: "memory")
#endif

#define NA   1024
#define NB   1024
#define NEGV (-1.0e20f)

#if HAVE_ASYNC_LDS
__device__ __forceinline__ void async_copy16(const float* gsrc, float* ldst) {
  // global_load_async_to_lds_b128: 16 bytes, tracked by ASYNCcnt.
  // Address-space casts go through integers: AS1 is 64-bit, AS3 is the 32-bit
  // LDS offset (low 32 bits of the generic shared pointer).
  __builtin_amdgcn_global_load_async_to_lds_b128(
      (gv4i_p)(uintptr_t)gsrc,
      (lv4i_p)(uint32_t)(uintptr_t)ldst,
      /*offset=*/0, /*cpol=*/0);
}
#endif

__global__ __launch_bounds__(NB)
void bpda_mu_softmax_kernel(const float* __restrict__ W,
                            const unsigned char* __restrict__ mask,
                            float* __restrict__ out) {
  __shared__ float mu[2][NB + 8];   // DP rows: entries 0..NB (1025 used)
  __shared__ float wbuf[2][NB];     // async-staged W rows (4 KB each)

  const int tid = threadIdx.x;          // column j = tid for output
  const int b   = blockIdx.x;

  const float* __restrict__ Wb = W + (size_t)b * NA * NB;
  const unsigned char* __restrict__ Mb = mask + (size_t)b * NA * NB;
  float* __restrict__ Ob = out + (size_t)b * NA * NB * 2;

  // ---- row 0 of mu: mu[0][0] = 0, else NEG -------------------------------
  mu[0][tid] = (tid == 0) ? 0.0f : NEGV;
  if (tid == 0) mu[0][NB] = NEGV;

  // ---- stage W row 0 -----------------------------------------------------
#if HAVE_ASYNC_LDS
  if (tid < NB / 4) {                     // waves 0..7 drive the DMA
    async_copy16(Wb + tid * 4, &wbuf[0][tid * 4]);
    WAIT_ASYNC0();
  }
#endif
  __syncthreads();

  int cur = 0;
  for (int i = 0; i < NA; ++i) {
    const int nxt = cur ^ 1;

#if HAVE_ASYNC_LDS
    // Kick off next row's DMA before touching this row: overlaps the whole
    // row of compute + the barrier. Buffer 'nxt' was last read before the
    // previous barrier, so no WAR hazard.
    if ((i + 1 < NA) && (tid < NB / 4)) {
      async_copy16(Wb + (size_t)(i + 1) * NB + tid * 4, &wbuf[nxt][tid * 4]);
    }
    const float w = wbuf[cur][tid];
#else
    const float w = __builtin_nontemporal_load(Wb + (size_t)i * NB + tid);
#endif

    const float a0 = mu[cur][tid];        // mu[i][j]
    const float a1 = mu[cur][tid + 1];    // mu[i][j+1]
    const float mk = (float)__builtin_nontemporal_load(Mb + (size_t)i * NB + tid);

    // Shared transcendental: t = exp(-|d|) feeds both softmax and logaddexp.
    const float d  = a1 - a0;
    const float t  = __expf(-__builtin_fabsf(d));        // v_exp_f32
    const float r  = __builtin_amdgcn_rcpf(1.0f + t);    // v_rcp_f32
    const float ph = r;                                  // prob of larger logit
    const float pl = t * r;
    const float p0 = (d >= 0.0f) ? ph : pl;              // softmax[.,0] (top)
    const float p1 = (d >= 0.0f) ? pl : ph;              // softmax[.,1] (top-left)

    v2f pv = { p0 * mk, p1 * mk };
    __builtin_nontemporal_store(pv, (v2f*)(Ob + ((size_t)i * NB + tid) * 2));

    // DP update: mu[i+1][tid+1] = logaddexp(a1, a0) + W[i][tid]
    const float lad = __builtin_fmaxf(a0, a1) + __logf(1.0f + t);  // v_log_f32
    mu[nxt][tid + 1] = lad + w;
    if (tid == 0) mu[nxt][0] = NEGV;

#if HAVE_ASYNC_LDS
    if (tid < NB / 4) WAIT_ASYNC0();      // loader waves: next W row landed
#endif
    __syncthreads();                      // publishes mu[nxt] (+ wbuf[nxt])
    cur = nxt;
  }
}

extern "C" void kernel_launch(void* const* d_in, const int* in_sizes, int n_in,
                              void* d_out, int out_size, void* d_ws, size_t ws_size,
                              hipStream_t stream) {
  (void)n_in; (void)out_size; (void)d_ws; (void)ws_size;
  const float* W = (const float*)d_in[0];
  const unsigned char* mask = (const unsigned char*)d_in[1];
  float* out = (float*)d_out;

  int B = in_sizes[0] / (NA * NB);
  if (B < 1) B = 1;

  hipLaunchKernelGGL(bpda_mu_softmax_kernel,
                     dim3(B), dim3(NB), 0, stream,
                     W, mask, out);
}